// CppnPotentialCA_53163105189978
// MI455X (gfx1250) — compile-verified
//
#include <hip/hip_runtime.h>
#include <math.h>

typedef float v2f __attribute__((ext_vector_type(2)));
typedef float v8f __attribute__((ext_vector_type(8)));

#define SZ 96
#define CH 9
#define NP 64
#define KT 125   // 5*5*5 taps
#define NIT 32   // 128 padded taps / K=4

// ---- Kernel 1: normalize kernels and emit B fragments in per-lane WMMA layout ----
// bw layout: [group g (8)][iter i (32)][lane (32)][2 floats] = 16384 floats (64 KB)
// lane: N = lane&15 (filter column), kbase = (lane>>4)*2 (K rows {0,1} or {2,3})
// value = (N<8 && tap<125) ? kn[8g+N][4i+kbase+j] : 0   -> zero-padding baked in
__global__ __launch_bounds__(256) void prep_bfrag(const float* __restrict__ k,
                                                  float* __restrict__ bw) {
    __shared__ float scale[NP];
    const int tid = threadIdx.x;
    if (tid < NP) {
        float sum = 0.f;
        for (int t = 0; t < KT; ++t) sum += k[tid * KT + t];
        scale[tid] = (sum > 0.f) ? (1.f / sum) : 1.f;
    }
    __syncthreads();
    for (int e = tid; e < 8 * NIT * 32; e += 256) {
        int g     = e >> 10;
        int i     = (e >> 5) & 31;
        int lane  = e & 31;
        int N     = lane & 15;
        int kbase = (lane >> 4) << 1;
        int t0    = 4 * i + kbase;
        int p     = g * 8 + (N & 7);
        float v0 = 0.f, v1 = 0.f;
        if (N < 8) {
            if (t0 < KT)     v0 = k[p * KT + t0] * scale[p];
            if (t0 + 1 < KT) v1 = k[p * KT + t0 + 1] * scale[p];
        }
        bw[e * 2]     = v0;
        bw[e * 2 + 1] = v1;
    }
}

// ---- Kernel 2: one block = 16 voxels along x at fixed (y,z); 8 waves = 8 source groups ----
__global__ __launch_bounds__(256) void ca_step(
    const float* __restrict__ x, const float* __restrict__ bw,
    const float* __restrict__ mArr, const float* __restrict__ sArr,
    const float* __restrict__ Tp, const int* __restrict__ c0,
    const int* __restrict__ c1, float* __restrict__ out)
{
    __shared__ float patch[8 * 500];    // [srcch][dz(5)][dy(5)][dx(20)]
    __shared__ float field[16 * 8];     // [voxel][cout-1]
    __shared__ int   offA[4 * NIT];     // tap -> patch offset (pairs load as b64)

    const int tid = threadIdx.x;
    const int x0  = blockIdx.x * 16;
    const int y   = blockIdx.y;
    const int z   = blockIdx.z;

    // tap -> offset table (taps >=125 clamp; their B weights are zero)
    if (tid < 4 * NIT) {
        int t = (tid < KT) ? tid : (KT - 1);
        int dz = t / 25, r = t - dz * 25, dy = r / 5, dx = r - dy * 5;
        offA[tid] = dz * 100 + dy * 20 + dx;
    }
    if (tid < 128) field[tid] = 0.f;
    // stage wrap-padded input halo for channels 1..8
    for (int i = tid; i < 8 * 500; i += 256) {
        int ch8 = i / 500;
        int r   = i - ch8 * 500;
        int dz = r / 100; int r2 = r - dz * 100;
        int dy = r2 / 20; int dx = r2 - dy * 20;
        int gz = z  - 2 + dz; if (gz < 0) gz += SZ; if (gz >= SZ) gz -= SZ;
        int gy = y  - 2 + dy; if (gy < 0) gy += SZ; if (gy >= SZ) gy -= SZ;
        int gx = x0 - 2 + dx; if (gx < 0) gx += SZ; if (gx >= SZ) gx -= SZ;
        patch[i] = x[(((gz * SZ) + gy) * SZ + gx) * CH + (ch8 + 1)];
    }
    __syncthreads();

    const int lane  = tid & 31;
    const int g     = tid >> 5;              // source group 0..7 (filters 8g..8g+7)
    const int mvox  = lane & 15;             // A: M row = voxel in tile
    const int kbase = (lane >> 4) << 1;      // A/B: K rows {0,1} or {2,3}
    const int N     = lane & 15;             // B/D: filter column
    int srcc = (c0[g * 8] - 1) & 7;          // source channel of this group
    const int pbase = srcc * 500 + mvox;

    // per-lane B-fragment stream: consecutive iters are 64 floats apart
    const float* bwg = bw + (g * NIT * 32 + lane) * 2;
    __builtin_prefetch(bwg, 0, 3);           // global_prefetch_b8

    v8f acc = {};
    #pragma unroll 4
    for (int i = 0; i < NIT; ++i) {
        int t0 = 4 * i + kbase;
        v2f a, b;
        a.x = patch[pbase + offA[t0]];
        a.y = patch[pbase + offA[t0 + 1]];
        b   = *(const v2f*)(bwg + i * 64);   // global_load_b64, immediate offset
        // D(16x16 f32) += A(16x4 f32) x B(4x16 f32)
        acc = __builtin_amdgcn_wmma_f32_16x16x4_f32(
            false, a, false, b, (short)0, acc, false, false);
    }

    // growth nonlinearity + segment-sum over c1 via LDS atomics
    if (N < 8) {
        int p = g * 8 + N;
        float pm = mArr[p];
        float ps = sArr[p];
        int cout = (c1[p] - 1) & 7;
        float inv2s2 = __builtin_amdgcn_rcpf(2.f * ps * ps);  // v_rcp_f32
        int vbase = (lane < 16) ? 0 : 8;     // C/D layout: VGPR r -> M = vbase + r
        #pragma unroll
        for (int r = 0; r < 8; ++r) {
            float d  = acc[r] - pm;
            float gr = __expf(-d * d * inv2s2) * 2.f - 1.f;
            atomicAdd(&field[(vbase + r) * 8 + cout], gr);
        }
    }
    __syncthreads();

    // final clipped Euler update: 16 voxels x 9 channels
    if (tid < 16 * CH) {
        int v = tid / CH, c = tid - v * CH;
        int gx  = x0 + v;
        int idx = (((z * SZ) + y) * SZ + gx) * CH + c;
        float f = (c == 0) ? 0.f : field[v * 8 + (c - 1)];
        float o = x[idx] + f / Tp[0];
        o = fminf(fmaxf(o, 0.f), 10.f);
        out[idx] = o;
    }
}

extern "C" void kernel_launch(void* const* d_in, const int* in_sizes, int n_in,
                              void* d_out, int out_size, void* d_ws, size_t ws_size,
                              hipStream_t stream) {
    (void)in_sizes; (void)n_in; (void)out_size; (void)ws_size;
    const float* x  = (const float*)d_in[0];
    const float* kk = (const float*)d_in[1];
    const float* m  = (const float*)d_in[2];
    const float* s  = (const float*)d_in[3];
    const float* T  = (const float*)d_in[4];
    const int*   c0 = (const int*)d_in[5];
    const int*   c1 = (const int*)d_in[6];
    float* bw = (float*)d_ws;  // 8*32*32*2 floats = 64 KB scratch

    prep_bfrag<<<1, 256, 0, stream>>>(kk, bw);
    dim3 grid(SZ / 16, SZ, SZ);
    ca_step<<<grid, 256, 0, stream>>>(x, bw, m, s, T, c0, c1, (float*)d_out);
}